// RelativePositionAttention_46420006535899
// MI455X (gfx1250) — compile-verified
//
#include <hip/hip_runtime.h>

#define B 2
#define L 2048
#define D 64
#define NREL 257          // 2*MAX_REL + 1
#define SCALE 0.125f      // 1/sqrt(64)

typedef __attribute__((ext_vector_type(2))) float v2f;
typedef __attribute__((ext_vector_type(4))) float v4f;
typedef __attribute__((ext_vector_type(8))) float v8f;

// ---------------------------------------------------------------------------
// Pass 1: p[b,i,t] = (q[b,i,:] . rel_pos_table[t,:]) * SCALE   (b*L*257 dots)
// ---------------------------------------------------------------------------
__global__ void relproj_kernel(const float* __restrict__ q,
                               const float* __restrict__ tab,
                               float* __restrict__ p) {
    int tid = blockIdx.x * blockDim.x + threadIdx.x;
    if (tid >= B * L * NREL) return;
    int t  = tid % NREL;
    int bi = tid / NREL;
    const v4f* qr = (const v4f*)(q + (size_t)bi * D);
    const v4f* tr = (const v4f*)(tab + (size_t)t * D);
    float s = 0.f;
#pragma unroll
    for (int d = 0; d < D / 4; ++d) {
        v4f a = qr[d];
        v4f b = tr[d];
        s += a.x * b.x + a.y * b.y + a.z * b.z + a.w * b.w;
    }
    p[tid] = s * SCALE;
}

// ---------------------------------------------------------------------------
// Pass 2: scores[b,i,j] = (q.k)/8 + p[b,i,clip(j-i+128)]  via f32 WMMA
// One wave per 16x16 tile; chain D/4 = 16 x V_WMMA_F32_16X16X4_F32.
// ---------------------------------------------------------------------------
__global__ void scores_kernel(const float* __restrict__ q,
                              const float* __restrict__ k,
                              const float* __restrict__ p,
                              float* __restrict__ W) {
    const int lane = threadIdx.x;       // 0..31, no divergence before WMMA
    const int m16  = lane & 15;
    const int h    = lane >> 4;         // k-half select (A 16x4 f32 layout)
    const int j0   = blockIdx.x * 16;
    const int i0   = blockIdx.y * 16;
    const int b    = blockIdx.z;

    const float* Q = q + ((size_t)b * L + i0) * D;  // rows i0..i0+15
    const float* K = k + ((size_t)b * L + j0) * D;  // rows j0..j0+15 (B = K^T)

    v8f acc = {};
#pragma unroll
    for (int kk = 0; kk < D; kk += 4) {
        const int c = kk + 2 * h;
        v2f a  = *(const v2f*)(Q + (size_t)m16 * D + c);  // A[m, c..c+1]
        v2f bb = *(const v2f*)(K + (size_t)m16 * D + c);  // B[c..c+1, n] = K[n, c..c+1]
        acc = __builtin_amdgcn_wmma_f32_16x16x4_f32(
            false, a, false, bb, (short)0, acc, false, false);
    }

    const float* pb = p + (size_t)b * L * NREL;
    float* Wb = W + (size_t)b * L * L;
#pragma unroll
    for (int r = 0; r < 8; ++r) {
        const int i = i0 + r + 8 * h;   // C/D layout: m = r + 8*(lane/16)
        const int j = j0 + m16;         //             n = lane%16
        int idx = j - i + 128;
        idx = idx < 0 ? 0 : (idx > 256 ? 256 : idx);
        Wb[(size_t)i * L + j] = acc[r] * SCALE + pb[(size_t)i * NREL + idx];
    }
}

// ---------------------------------------------------------------------------
// Pass 3: row-wise softmax in place over L=2048 (one 256-thread block / row)
// ---------------------------------------------------------------------------
__global__ void softmax_kernel(float* __restrict__ W) {
    __shared__ float red[256];
    const int tid = threadIdx.x;
    float* row = W + (size_t)blockIdx.x * L;

    float m = -3.402823466e38f;
    for (int j = tid; j < L; j += 256) m = fmaxf(m, row[j]);
    red[tid] = m;
    __syncthreads();
    for (int s = 128; s > 0; s >>= 1) {
        if (tid < s) red[tid] = fmaxf(red[tid], red[tid + s]);
        __syncthreads();
    }
    m = red[0];
    __syncthreads();

    float sum = 0.f;
    for (int j = tid; j < L; j += 256) {
        float e = __expf(row[j] - m);
        row[j] = e;
        sum += e;
    }
    red[tid] = sum;
    __syncthreads();
    for (int s = 128; s > 0; s >>= 1) {
        if (tid < s) red[tid] += red[tid + s];
        __syncthreads();
    }
    const float inv = 1.f / red[0];
    for (int j = tid; j < L; j += 256) row[j] *= inv;
}

// ---------------------------------------------------------------------------
// Pass 4: out[b,i,:] = weights[b,i,:] @ v[b,:,:]   via f32 WMMA
// One wave per 16(i) x 16(d) tile, K loop over L in steps of 4.
// ---------------------------------------------------------------------------
__global__ void av_kernel(const float* __restrict__ W,
                          const float* __restrict__ v,
                          float* __restrict__ out) {
    const int lane = threadIdx.x;
    const int m16  = lane & 15;
    const int h    = lane >> 4;
    const int n0   = blockIdx.x * 16;   // D tile (0 or 16 or 32 or 48)
    const int i0   = blockIdx.y * 16;
    const int b    = blockIdx.z;

    const float* Wr = W + ((size_t)b * L + i0) * L;
    const float* Vb = v + (size_t)b * L * D;

    v8f acc = {};
#pragma unroll 8
    for (int kk = 0; kk < L; kk += 4) {
        const int c = kk + 2 * h;
        v2f a = *(const v2f*)(Wr + (size_t)m16 * L + c);  // A = weights tile
        v2f bb;                                           // B[c, n] = V[c, n0+n]
        bb.x = Vb[(size_t)c * D + n0 + m16];
        bb.y = Vb[(size_t)(c + 1) * D + n0 + m16];
        acc = __builtin_amdgcn_wmma_f32_16x16x4_f32(
            false, a, false, bb, (short)0, acc, false, false);
    }
#pragma unroll
    for (int r = 0; r < 8; ++r) {
        const int i = i0 + r + 8 * h;
        out[((size_t)b * L + i) * D + n0 + m16] = acc[r];
    }
}

// ---------------------------------------------------------------------------
extern "C" void kernel_launch(void* const* d_in, const int* in_sizes, int n_in,
                              void* d_out, int out_size, void* d_ws, size_t ws_size,
                              hipStream_t stream) {
    (void)in_sizes; (void)n_in; (void)out_size; (void)ws_size;
    const float* q   = (const float*)d_in[0];
    const float* k   = (const float*)d_in[1];
    const float* v   = (const float*)d_in[2];
    const float* tab = (const float*)d_in[3];

    float* out = (float*)d_out;                       // [B, L, D]
    float* W   = out + (size_t)B * L * D;             // [B, L, L] (scores->weights)
    float* p   = (float*)d_ws;                        // [B, L, NREL], 4.2 MB

    const int np = B * L * NREL;
    relproj_kernel<<<(np + 255) / 256, 256, 0, stream>>>(q, tab, p);
    scores_kernel<<<dim3(L / 16, L / 16, B), 32, 0, stream>>>(q, k, p, W);
    softmax_kernel<<<B * L, 256, 0, stream>>>(W);
    av_kernel<<<dim3(D / 16, L / 16, B), 32, 0, stream>>>(W, v, out);
}